// MedianBlur_55439437857044
// MI455X (gfx1250) — compile-verified
//
#include <hip/hip_runtime.h>
#include <stdint.h>

// Median blur 5x5, zero padding, input (4,3,1024,1024) f32 -> same shape.
// Exact median via doubly-sorted 5x5 window reduced to its 13-candidate
// anti-diagonal band, then structured merges + 2-array selection identity.
// CDNA5 paths: wave32 blocks, async global->LDS tile loads (ASYNCcnt),
// s_wait_asynccnt, LDS with conflict-free padding, min3/max3-fusable VALU.

#define IMG_W     1024
#define IMG_H     1024
#define NPLANES   12            // B*C = 4*3
#define TILE_OW   64
#define TILE_OH   16
#define HALO      2
#define TILE_IW   (TILE_OW + 2 * HALO)   // 68
#define TILE_IH   (TILE_OH + 2 * HALO)   // 20
#define LDS_STRIDE 73                    // pad 68->73: conflict-free column reads
#define THREADS   256

__device__ __forceinline__ void ce(float& x, float& y) {
  float t = fminf(x, y);
  y = fmaxf(x, y);
  x = t;
}

// optimal 9-CE sorting network for 5 elements
__device__ __forceinline__ void sort5(float& a, float& b, float& c, float& d, float& e) {
  ce(a, b); ce(d, e); ce(c, e); ce(c, d); ce(a, d);
  ce(a, c); ce(b, e); ce(b, d); ce(b, c);
}

// A..E: 5 columns of the window, each sorted ascending. Returns exact median of 25.
__device__ __forceinline__ float med25(const float A[5], const float B[5],
                                       const float C[5], const float D[5],
                                       const float E[5]) {
  // Row sort (columns remain sorted -> doubly-sorted matrix).
  float r00 = A[0], r01 = B[0], r02 = C[0], r03 = D[0], r04 = E[0];
  float r10 = A[1], r11 = B[1], r12 = C[1], r13 = D[1], r14 = E[1];
  float r20 = A[2], r21 = B[2], r22 = C[2], r23 = D[2], r24 = E[2];
  float r30 = A[3], r31 = B[3], r32 = C[3], r33 = D[3], r34 = E[3];
  float r40 = A[4], r41 = B[4], r42 = C[4], r43 = D[4], r44 = E[4];
  sort5(r00, r01, r02, r03, r04);
  sort5(r10, r11, r12, r13, r14);
  sort5(r20, r21, r22, r23, r24);
  sort5(r30, r31, r32, r33, r34);
  sort5(r40, r41, r42, r43, r44);
  // 13-candidate anti-diagonal band (median of 25 == 7th smallest of these):
  float a = r03, b = r04;                 // row0: top-2
  float c = r12, d = r13, e = r14;        // row1: top-3
  float f = r21, g = r22, h = r23;        // row2: middle-3
  float i = r30, j = r31, k = r32;        // row3: bottom-3
  float l = r40, m = r41;                 // row4: bottom-2
  // S1 = sorted {a,b,d,e,h} (uses known a<=b<=e, a<=d<=h, d<=e): P=[a,b,d,h,e]
  ce(b, d); ce(d, h); ce(h, e);
  // S2 = sorted {i,f,l,j,m} (uses known i<=l<=m, f<=j<=m, i<=j): Q=[i,f,l,j,m]
  ce(i, f); ce(f, l); ce(l, j);
  // Batcher merge(P,Q), pruned to middle outputs T3..T6 of the sorted 10.
  // evens-of-evens merge22([P0,P4],[Q0,Q4]) -> w=[u0,v0,u1,v1]
  float u0 = a, u1 = e, v0 = i, v1 = m;
  ce(u0, v0); ce(u1, v1); ce(v0, u1);
  ce(d, l);                  // odds-of-evens: o=[d,l]
  ce(v0, d);                 // D1=v0, D2=d
  ce(u1, l);                 // D3=u1, D4=l
  ce(b, f); ce(h, j); ce(f, h);  // merge22 of odds: Eo=[b,f,h,j]
  ce(d, f);                  // T3=d, T4=f
  ce(u1, h);                 // T5=u1, T6=h
  // 7th of union(T[10], [c<=g<=k]) = min over splits of max pairs:
  float z0 = fminf(h, fmaxf(u1, c));
  float z1 = fminf(fmaxf(f, g), fmaxf(d, k));
  return fminf(z0, z1);
}

__global__ __launch_bounds__(THREADS)
void median5x5_kernel(const float* __restrict__ in, float* __restrict__ out) {
  __shared__ float tile[TILE_IH * LDS_STRIDE];

  const int tid   = threadIdx.x;
  const int plane = blockIdx.z;
  const int x0    = blockIdx.x * TILE_OW;
  const int y0    = blockIdx.y * TILE_OH;
  const float* pin = in + (size_t)plane * (IMG_W * IMG_H);

  // Zero-fill padded tile (supplies the zero padding at image borders).
  for (int t = tid; t < TILE_IH * LDS_STRIDE; t += THREADS) tile[t] = 0.0f;
  __syncthreads();

  // Async global -> LDS tile load (CDNA5 ASYNC path). Generic LDS pointer's
  // low 32 bits are the in-group LDS byte offset on gfx1250.
  const uint32_t lds_base = (uint32_t)(uintptr_t)(&tile[0]);
  for (int t = tid; t < TILE_IH * TILE_IW; t += THREADS) {
    int r  = t / TILE_IW;
    int cx = t - r * TILE_IW;
    int gx = x0 - HALO + cx;
    int gy = y0 - HALO + r;
    if ((unsigned)gx < (unsigned)IMG_W && (unsigned)gy < (unsigned)IMG_H) {
      const float* gp = pin + (size_t)gy * IMG_W + gx;
      uint32_t loff = lds_base + (uint32_t)(r * LDS_STRIDE + cx) * 4u;
      asm volatile("global_load_async_to_lds_b32 %0, %1, off"
                   :: "v"(loff), "v"(gp)
                   : "memory");
    }
  }
  asm volatile("s_wait_asynccnt 0" ::: "memory");
  __syncthreads();

  // Each thread: 4 consecutive outputs in one row; 8 column-sorts shared
  // across the 4 overlapping 5x5 windows.
  const int tx4 = (tid & 15) * 4;
  const int ty  = tid >> 4;

  float col[8][5];
#pragma unroll
  for (int cc = 0; cc < 8; ++cc) {
#pragma unroll
    for (int r = 0; r < 5; ++r)
      col[cc][r] = tile[(ty + r) * LDS_STRIDE + tx4 + cc];
    sort5(col[cc][0], col[cc][1], col[cc][2], col[cc][3], col[cc][4]);
  }

  float4 res;
  res.x = med25(col[0], col[1], col[2], col[3], col[4]);
  res.y = med25(col[1], col[2], col[3], col[4], col[5]);
  res.z = med25(col[2], col[3], col[4], col[5], col[6]);
  res.w = med25(col[3], col[4], col[5], col[6], col[7]);

  float* pout = out + (size_t)plane * (IMG_W * IMG_H)
                    + (size_t)(y0 + ty) * IMG_W + (x0 + tx4);
  *reinterpret_cast<float4*>(pout) = res;
}

extern "C" void kernel_launch(void* const* d_in, const int* in_sizes, int n_in,
                              void* d_out, int out_size, void* d_ws, size_t ws_size,
                              hipStream_t stream) {
  (void)in_sizes; (void)n_in; (void)d_ws; (void)ws_size; (void)out_size;
  const float* in = (const float*)d_in[0];
  float* out = (float*)d_out;
  dim3 grid(IMG_W / TILE_OW, IMG_H / TILE_OH, NPLANES);
  median5x5_kernel<<<grid, dim3(THREADS), 0, stream>>>(in, out);
}